// PosEvalModel_45758581571755
// MI455X (gfx1250) — compile-verified
//
#include <hip/hip_runtime.h>
#include <cstdint>

typedef float v2f  __attribute__((ext_vector_type(2)));
typedef float v8f  __attribute__((ext_vector_type(8)));
typedef uint32_t u32x4 __attribute__((ext_vector_type(4)));
typedef uint32_t u32x8 __attribute__((ext_vector_type(8)));

#define STATE 128
#define NCTX  64
#define BATCH 8192
#define BT    64            // batch rows per workgroup tile
#define RS    130           // padded LDS row stride (floats): 128 + 2 (TDM pad)
#define TILE_ELEMS (BT*STATE)   // 8192 f32 per tile (contiguous in global)

// ws layout (floats)
#define WS_CKI 0
#define WS_CQI 16384
#define WS_CV3 32768
#define WS_BQ  33152
#define WS_BK  33280
#define WS_BV3 33408

// ---------------------------------------------------------------------------
// Prologue: fold the linear chain into combined weights in d_ws.
//   Ck = Wk @ Wp'          (128x128)   k_c = in_c @ Ck^T + bk
//   Cq = Wq @ Wp'          (128x128)   q   = in_0 @ Cq^T + bq   (indicator col folded)
//   Cv3 = Wf @ Wv @ Wp'    (3x128)     vf_c = in_c @ Cv3^T + bv3
// Ck/Cq stored pre-swizzled for WMMA B-fragment float2 loads:
//   I[(k>>1)*256 + n*2 + (k&1)] = C[n][k]
// ---------------------------------------------------------------------------
__global__ void __launch_bounds__(256) pos_eval_prep(
    const float* __restrict__ projW, const float* __restrict__ projB,
    const float* __restrict__ qW,    const float* __restrict__ qB,
    const float* __restrict__ kvW,   const float* __restrict__ kvB,
    const float* __restrict__ fW,    float* __restrict__ ws)
{
  __shared__ float wfv[3*128];
  const int t = threadIdx.x;

  // Wfv = Wf @ Wv   (Wv = kvW rows 128..255)
  for (int idx = t; idx < 3*128; idx += 256) {
    int o = idx >> 7, i = idx & 127;
    float s = 0.f;
    for (int k = 0; k < 128; ++k) s += fW[o*128 + k] * kvW[(128 + k)*128 + i];
    wfv[idx] = s;
  }
  __syncthreads();

  float* CkI = ws + WS_CKI;
  float* CqI = ws + WS_CQI;
  float* Cv3 = ws + WS_CV3;
  float* bq  = ws + WS_BQ;
  float* bk  = ws + WS_BK;
  float* bv3 = ws + WS_BV3;

  for (int idx = t; idx < 16384; idx += 256) {
    int n = idx & 127, kd = idx >> 7;
    float sk = 0.f, sq = 0.f;
    for (int i = 0; i < 128; ++i) {
      float wp = projW[i*129 + kd];          // Wp'[i][kd]
      sk += kvW[n*128 + i] * wp;             // Wk row n
      sq += qW [n*128 + i] * wp;
    }
    int off = (kd >> 1)*256 + n*2 + (kd & 1);
    CkI[off] = sk;
    CqI[off] = sq;
  }

  for (int idx = t; idx < 384; idx += 256) {
    int o = idx >> 7, kd = idx & 127;
    float s = 0.f;
    for (int i = 0; i < 128; ++i) s += wfv[o*128 + i] * projW[i*129 + kd];
    Cv3[idx] = s;
  }

  if (t < 128) {
    float sb = 0.f, skb = 0.f;
    for (int i = 0; i < 128; ++i) {
      sb  += qW [t*128 + i] * (projB[i] + projW[i*129 + 128]); // +indicator col
      skb += kvW[t*128 + i] * projB[i];
    }
    bq[t] = sb  + qB[t];
    bk[t] = skb + kvB[t];
  }
  if (t < 3) {
    float s = 0.f;
    for (int i = 0; i < 128; ++i) s += wfv[t*128 + i] * projB[i];
    for (int k = 0; k < 128; ++k) s += fW[t*128 + k] * kvB[128 + k];
    bv3[t] = s;
  }
}

// ---------------------------------------------------------------------------
// TDM: issue a 32KB contiguous tile load into LDS, with 2-dword padding every
// 128 dwords so LDS rows land at stride 130 (bank-conflict-free vf stage).
// ---------------------------------------------------------------------------
__device__ __forceinline__ uint32_t lds_off(const void* p) {
  return (uint32_t)(uintptr_t)p;               // low 32 bits of flat LDS addr = offset
}

__device__ __forceinline__ void tdm_load_tile(const float* gsrc, uint32_t ldsaddr) {
  uint64_t ga = (uint64_t)(uintptr_t)gsrc;
  u32x4 g0;
  g0.x = 1u;                                   // count=1, user descriptor
  g0.y = ldsaddr;                              // lds_addr
  g0.z = (uint32_t)ga;                         // global_addr[31:0]
  g0.w = (uint32_t)(ga >> 32) | (2u << 30);    // global_addr[56:32] | type=2 (image)
  u32x8 g1;
  g1[0] = 0x00020000u                          // data_size=2 (4B), wg_mask=0
        | (1u << 20)                           // pad_enable
        | (6u << 22)                           // pad_interval: every 128 dwords
        | (1u << 25);                          // pad_amount: 2 dwords
  g1[1] = (uint32_t)(TILE_ELEMS & 0xFFFF) << 16;   // tensor_dim0 lo16
  g1[2] = (uint32_t)(TILE_ELEMS >> 16) | (1u << 16); // tensor_dim0 hi16 | tensor_dim1=1
  g1[3] = (uint32_t)(TILE_ELEMS & 0xFFFF) << 16;   // tile_dim0 = 8192 (1D tile)
  g1[4] = 0u;                                  // tile_dim1=0 (unused), tile_dim2=0
  g1[5] = (uint32_t)TILE_ELEMS;                // tensor_dim0_stride lo32
  g1[6] = 0u;
  g1[7] = 0u;
  asm volatile("tensor_load_to_lds %0, %1" :: "s"(g0), "s"(g1) : "memory");
}

__device__ __forceinline__ float sigmoidf_(float x) {
  return 1.0f / (1.0f + __expf(-x));
}

// ---------------------------------------------------------------------------
// Main fused kernel: one WG per 64-batch tile (128 WGs), 8 waves.
// ---------------------------------------------------------------------------
__global__ void __launch_bounds__(256, 1) pos_eval_main(
    const float* __restrict__ tacst, const float* __restrict__ ws,
    const float* __restrict__ finalB, float* __restrict__ out)
{
  __shared__ float combI[128*128];   // 64KB: CqI then CkI (swizzled B-matrix)
  __shared__ float inb[2][BT*RS];    // double-buffered TDM input tiles (padded rows)
  __shared__ float cv3s[3*128];
  __shared__ float bks[128];
  __shared__ float red[BT*2];        // per-row partial k.q dots (2 waves per strip)
  __shared__ float gates[BT];

  const int t     = threadIdx.x;
  const int lane  = t & 31;
  const int l     = lane & 15;       // 16-lane half index -> N / M selector
  const int h     = lane >> 4;       // half id
  const int w     = t >> 5;          // wave 0..7
  const int strip = w >> 1;          // row strip: rows [16*strip, 16*strip+15]
  const int m0    = strip * 16;
  const int n0b   = (w & 1) * 64;    // wave's 64-column slab
  const size_t b0 = (size_t)blockIdx.x * BT;

  const float* CkI  = ws + WS_CKI;
  const float* CqI  = ws + WS_CQI;
  const float* Cv3g = ws + WS_CV3;
  const float* bqg  = ws + WS_BQ;
  const float* bkg  = ws + WS_BK;
  const float* bv3g = ws + WS_BV3;

  // Stage small constants + CqI into LDS; kick off TDM for tile c=0.
  for (int i = t; i < 384; i += 256) cv3s[i] = Cv3g[i];
  if (t < 128) bks[t] = bkg[t];
  for (int i = t; i < 16384; i += 256) combI[i] = CqI[i];
  if (w == 0) tdm_load_tile(tacst + b0*STATE, lds_off(&inb[0][0]));
  if (w == 0) __builtin_amdgcn_s_wait_tensorcnt(0);
  __syncthreads();                              // tile0 + CqI visible
  if (w == 0) tdm_load_tile(tacst + ((size_t)BATCH + b0)*STATE, lds_off(&inb[1][0]));

  // ---- q = in0 @ Cq^T + bq, kept as per-wave D-fragments in VGPRs ----
  v8f qf[4];
  #pragma unroll
  for (int tt = 0; tt < 4; ++tt) {
    v8f acc = {0.f,0.f,0.f,0.f,0.f,0.f,0.f,0.f};
    const int n0 = n0b + tt*16;
    #pragma unroll
    for (int s = 0; s < 32; ++s) {
      const int ks = 4*s + 2*h;                                 // even
      v2f a = *(const v2f*)&inb[0][(m0 + l)*RS + ks];           // A[m, ks..ks+1]
      v2f b = *(const v2f*)&combI[(2*s + h)*256 + (n0 + l)*2];  // B[ks..ks+1, n]
      acc = __builtin_amdgcn_wmma_f32_16x16x4_f32(
          false, a, false, b, (short)0, acc, false, false);
    }
    float bqv = bqg[n0 + l];
    #pragma unroll
    for (int v = 0; v < 8; ++v) acc[v] += bqv;
    qf[tt] = acc;
  }
  __syncthreads();                              // q done reading combI
  for (int i = t; i < 16384; i += 256) combI[i] = CkI[i];   // swap in CkI

  float accO = 0.f;
  const int ro = t & 63;                        // vf-stage row
  const int oo = t >> 6;                        // vf-stage output col (0..2 active)
  const float bv3v = (oo < 3) ? bv3g[oo] : 0.f;
  __syncthreads();                              // CkI ready

  for (int c = 1; c < NCTX; ++c) {
    const int buf = c & 1;
    if (w == 0) __builtin_amdgcn_s_wait_tensorcnt(0);
    __syncthreads();                            // tile c ready, in[buf^1] free
    if (w == 0 && c + 1 < NCTX)
      tdm_load_tile(tacst + ((size_t)(c + 1)*BATCH + b0)*STATE,
                    lds_off(&inb[buf ^ 1][0]));

    const float* ib = &inb[buf][0];

    // ---- k = in_c @ Ck^T + bk ; fused dot with resident q fragments ----
    float pv[8];
    #pragma unroll
    for (int v = 0; v < 8; ++v) pv[v] = 0.f;
    #pragma unroll
    for (int tt = 0; tt < 4; ++tt) {
      v8f acc = {0.f,0.f,0.f,0.f,0.f,0.f,0.f,0.f};
      const int n0 = n0b + tt*16;
      #pragma unroll
      for (int s = 0; s < 32; ++s) {
        const int ks = 4*s + 2*h;
        v2f a = *(const v2f*)&ib[(m0 + l)*RS + ks];
        v2f b = *(const v2f*)&combI[(2*s + h)*256 + (n0 + l)*2];
        acc = __builtin_amdgcn_wmma_f32_16x16x4_f32(
            false, a, false, b, (short)0, acc, false, false);
      }
      float bkv = bks[n0 + l];
      #pragma unroll
      for (int v = 0; v < 8; ++v) pv[v] += (acc[v] + bkv) * qf[tt][v];
    }
    // reduce over the wave's 16 N-lanes (keeps the two halves separate)
    #pragma unroll
    for (int v = 0; v < 8; ++v) {
      pv[v] += __shfl_xor(pv[v], 1, 32);
      pv[v] += __shfl_xor(pv[v], 2, 32);
      pv[v] += __shfl_xor(pv[v], 4, 32);
      pv[v] += __shfl_xor(pv[v], 8, 32);
    }
    if (l == 0) {                                // lanes 0 (h=0) and 16 (h=1)
      const int base = m0 + 8*h;
      #pragma unroll
      for (int v = 0; v < 8; ++v) red[(base + v)*2 + (w & 1)] = pv[v];
    }
    __syncthreads();
    if (t < BT)
      gates[t] = sigmoidf_((red[2*t] + red[2*t + 1]) * 0.08838834764831845f);
    __syncthreads();

    // ---- vf_c = in_c @ Cv3^T + bv3 (3 cols); accumulate gated output ----
    if (oo < 3) {
      float dd = 0.f;
      #pragma unroll 8
      for (int s = 0; s < 128; ++s) dd += ib[ro*RS + s] * cv3s[oo*128 + s];
      accO += gates[ro] * (dd + bv3v);
    }
  }

  if (oo < 3) out[(b0 + ro)*3 + oo] = accO + finalB[oo];
}

// ---------------------------------------------------------------------------
extern "C" void kernel_launch(void* const* d_in, const int* in_sizes, int n_in,
                              void* d_out, int out_size, void* d_ws, size_t ws_size,
                              hipStream_t stream) {
  (void)in_sizes; (void)n_in; (void)out_size; (void)ws_size;
  const float* tacst = (const float*)d_in[0];
  const float* projW = (const float*)d_in[1];
  const float* projB = (const float*)d_in[2];
  const float* qW    = (const float*)d_in[3];
  const float* qB    = (const float*)d_in[4];
  const float* kvW   = (const float*)d_in[5];
  const float* kvB   = (const float*)d_in[6];
  // d_in[7] = attn_sv_init (zeros) -- folded out
  const float* fW    = (const float*)d_in[8];
  const float* fB    = (const float*)d_in[9];
  float* ws = (float*)d_ws;

  pos_eval_prep<<<1, 256, 0, stream>>>(projW, projB, qW, qB, kvW, kvB, fW, ws);
  pos_eval_main<<<BATCH/BT, 256, 0, stream>>>(tacst, ws, fB, (float*)d_out);
}